// AxialAttention_gated_data_5171140624845
// MI455X (gfx1250) — compile-verified
//
#include <hip/hip_runtime.h>
#include <hip/hip_bf16.h>

// ---------------------------------------------------------------------------
// AxialAttention (gated) for MI455X / gfx1250.
// fp32 end-to-end; matrix contractions use V_WMMA_F32_16X16X4_F32 (wave32).
// Problem is bandwidth-bound (~8.6 GFLOP vs ~400MB traffic; fits in 192MB L2),
// so fp32 WMMA keeps reference precision at no cost.
// ---------------------------------------------------------------------------

typedef __attribute__((ext_vector_type(2))) float v2f;
typedef __attribute__((ext_vector_type(8))) float v8f;

__device__ __forceinline__ v8f wmma4(v2f a, v2f b, v8f c) {
  // D(16x16) = A(16x4) * B(4x16) + C, all fp32.
  return __builtin_amdgcn_wmma_f32_16x16x4_f32(false, a, false, b, (short)0, c,
                                               false, false);
}

static constexpr int N_  = 16;
static constexpr int C_  = 128;
static constexpr int H_  = 64;
static constexpr int W_  = 64;
static constexpr int B_  = N_ * W_;   // 1024
static constexpr int OC_ = 256;       // OUT_PLANES*2
static constexpr float EPS_ = 1e-5f;

// ---- kernel 1: xn[b,c] = mean_h x[n,c,h,w],  b = n*64 + w -----------------
__global__ void k_colmean(const float* __restrict__ x, float* __restrict__ xn) {
  int nc = blockIdx.x;            // n*128 + c
  int n = nc >> 7, c = nc & 127;
  int w = threadIdx.x;            // 64 threads
  const float* p = x + (size_t)nc * 4096 + w;   // (nc*64 + h)*64 + w
  float s = 0.f;
  for (int h = 0; h < 64; ++h) s += p[h * 64];
  xn[(size_t)(n * 64 + w) * 128 + c] = s * (1.f / 64.f);
}

// ---- kernel 2: gate MLP -> sigmoid gates (B,4) ----------------------------
__global__ void k_gate(const float* __restrict__ xn,
                       const float* __restrict__ w1, const float* __restrict__ b1,
                       const float* __restrict__ w2, const float* __restrict__ b2,
                       float* __restrict__ sig) {
  int b = blockIdx.x;
  int t = threadIdx.x;            // 128 threads
  __shared__ float sx[128];
  __shared__ float sh1[128];
  sx[t] = xn[(size_t)b * 128 + t];
  __syncthreads();
  float acc = b1[t];
  for (int c = 0; c < 128; ++c) acc = fmaf(w1[t * 128 + c], sx[c], acc);
  sh1[t] = fmaxf(acc, 0.f);
  __syncthreads();
  if (t < 4) {
    float a2 = b2[t];
    for (int c = 0; c < 128; ++c) a2 = fmaf(w2[t * 128 + c], sh1[c], a2);
    a2 = fmaxf(a2, 0.f);
    sig[b * 4 + t] = 1.f / (1.f + expf(-a2));
  }
}

// ---- kernel 3: gather relative embeddings: emb[c][i][j] = rel[c][i-j+63] --
__global__ void k_emb(const float* __restrict__ rel, float* __restrict__ emb) {
  int blk = blockIdx.x;           // c*64 + i
  int c = blk >> 6, i = blk & 63;
  int j = threadIdx.x;            // 64 threads
  emb[(size_t)blk * 64 + j] = rel[c * 127 + (i - j + 63)];
}

// ---- kernel 4: QKV GEMM:  qkv[b][o][h] = sum_c w_qkv[o][c] * x[n][c][h][w] -
// One wave per 16(o) x 16(w) tile at fixed (n,h); K = 128 via 32 WMMA steps.
__global__ void k_qkv_gemm(const float* __restrict__ x,
                           const float* __restrict__ wq,
                           float* __restrict__ qkv) {
  int wid  = blockIdx.x * 8 + (threadIdx.x >> 5);  // 65536 waves
  int lane = threadIdx.x & 31;
  int ot = wid & 15;              // o-tile   (16)
  int wt = (wid >> 4) & 3;        // w-tile   (4)
  int h  = (wid >> 6) & 63;       // h        (64)
  int n  = wid >> 12;             // n        (16)
  int o0 = ot * 16, w0 = wt * 16;
  int row = lane & 15;
  int kb  = (lane >> 4) << 1;     // 0 or 2

  const float* arow = wq + (size_t)(o0 + row) * 128 + kb;
  const float* bcol = x + (((size_t)(n * 128 + kb) * 64 + h) * 64) + w0 + row;

  v8f acc = {0.f, 0.f, 0.f, 0.f, 0.f, 0.f, 0.f, 0.f};
  for (int k = 0; k < 32; ++k) {
    v2f a, b;
    a.x = arow[k * 4 + 0];
    a.y = arow[k * 4 + 1];
    b.x = bcol[(size_t)k * 16384];          // c = k*4+kb
    b.y = bcol[(size_t)k * 16384 + 4096];   // c = k*4+kb+1
    acc = wmma4(a, b, acc);
  }
  int col = lane & 15;
  int mb  = (lane >> 4) * 8;
  int bidx = n * 64 + (w0 + col);
  float* dst = qkv + ((size_t)bidx * 256 + o0 + mb) * 64 + h;
#pragma unroll
  for (int r = 0; r < 8; ++r) dst[r * 64] = acc[r];
}

// ---- kernel 5/10: per-channel BN stats over [b][ch][64] -------------------
__global__ void k_chan_stats(const float* __restrict__ data,
                             const float* __restrict__ g, const float* __restrict__ bt,
                             float* __restrict__ scale, float* __restrict__ shift) {
  int o = blockIdx.x;             // 256 channels
  int t = threadIdx.x;            // 256 threads
  float s = 0.f, q = 0.f;
  for (int idx = t; idx < B_ * 64; idx += 256) {
    int b = idx >> 6, h = idx & 63;
    float v = data[((size_t)b * 256 + o) * 64 + h];
    s += v; q += v * v;
  }
  __shared__ float rs[256], rq[256];
  rs[t] = s; rq[t] = q;
  __syncthreads();
  for (int st = 128; st > 0; st >>= 1) {
    if (t < st) { rs[t] += rs[t + st]; rq[t] += rq[t + st]; }
    __syncthreads();
  }
  if (t == 0) {
    const float cnt = (float)(B_ * 64);
    float mean = rs[0] / cnt;
    float var  = rq[0] / cnt - mean * mean;
    float sc   = g[o] * rsqrtf(var + EPS_);
    scale[o] = sc;
    shift[o] = bt[o] - mean * sc;
  }
}

// ---- kernel 6: per-(b,g) compute qk/qr/kr, emit partial stats -------------
__global__ void k_sim_stats(const float* __restrict__ qkv,
                            const float* __restrict__ qsc, const float* __restrict__ qsh,
                            const float* __restrict__ emb,
                            const float* __restrict__ sig,
                            float* __restrict__ part) {
  int bg = blockIdx.x;            // b*8 + g
  int g = bg & 7, b = bg >> 3;
  int tid = threadIdx.x;          // 256 threads
  __shared__ float ql[8][64], kl[8][64];
  for (int idx = tid; idx < 1024; idx += 256) {
    int c = idx >> 6, hh = idx & 63;
    int o = g * 32 + c;
    float v = qsc[o] * qkv[((size_t)b * 256 + o) * 64 + hh] + qsh[o];
    if (c < 8) ql[c][hh] = v; else kl[c - 8][hh] = v;
  }
  __syncthreads();

  int wid = tid >> 5, lane = tid & 31;
  int row = lane & 15, kb = (lane >> 4) << 1;
  float s_qk = 0.f, q_qk = 0.f;
  for (int tt = 0; tt < 2; ++tt) {
    int tile = wid + tt * 8;      // 16 tiles over 8 waves
    int i0 = (tile >> 2) * 16, j0 = (tile & 3) * 16;
    v8f acc = {0.f, 0.f, 0.f, 0.f, 0.f, 0.f, 0.f, 0.f};
    for (int ks = 0; ks < 8; ks += 4) {
      v2f a, bb;
      a.x  = ql[ks + kb + 0][i0 + row];
      a.y  = ql[ks + kb + 1][i0 + row];
      bb.x = kl[ks + kb + 0][j0 + row];
      bb.y = kl[ks + kb + 1][j0 + row];
      acc = wmma4(a, bb, acc);
    }
#pragma unroll
    for (int r = 0; r < 8; ++r) { float v = acc[r]; s_qk += v; q_qk += v * v; }
  }

  float sig1 = sig[b * 4 + 0], sig2 = sig[b * 4 + 1];
  float s_qr = 0.f, q_qr = 0.f, s_kr = 0.f, q_kr = 0.f;
  for (int p = tid; p < 4096; p += 256) {
    int i = p >> 6, j = p & 63;
    float aq = 0.f, ak = 0.f;
#pragma unroll
    for (int c = 0; c < 8; ++c) {
      aq = fmaf(ql[c][i], emb[((size_t)(c * 64 + i)) * 64 + j], aq);
      ak = fmaf(kl[c][j], emb[((size_t)((8 + c) * 64 + j)) * 64 + i], ak);
    }
    aq *= sig1; ak *= sig2;
    s_qr += aq; q_qr += aq * aq; s_kr += ak; q_kr += ak * ak;
  }

  __shared__ float red[256];
  float vals[6] = {s_qk, q_qk, s_qr, q_qr, s_kr, q_kr};
  for (int v = 0; v < 6; ++v) {
    red[tid] = vals[v];
    __syncthreads();
    for (int st = 128; st > 0; st >>= 1) {
      if (tid < st) red[tid] += red[tid + st];
      __syncthreads();
    }
    if (tid == 0) part[(size_t)bg * 6 + v] = red[0];
    __syncthreads();
  }
}

// ---- kernel 7: finalize sim BN (24 channels), deterministic reduce --------
__global__ void k_sim_final(const float* __restrict__ part,
                            const float* __restrict__ g_sim, const float* __restrict__ b_sim,
                            float* __restrict__ ssc, float* __restrict__ ssh) {
  int ch = blockIdx.x;            // 24 channels: type*8 + g
  int type = ch >> 3, g = ch & 7;
  int t = threadIdx.x;            // 256 threads
  float s = 0.f, q = 0.f;
  for (int b = t; b < B_; b += 256) {
    s += part[(size_t)(b * 8 + g) * 6 + type * 2 + 0];
    q += part[(size_t)(b * 8 + g) * 6 + type * 2 + 1];
  }
  __shared__ float rs[256], rq[256];
  rs[t] = s; rq[t] = q;
  __syncthreads();
  for (int st = 128; st > 0; st >>= 1) {
    if (t < st) { rs[t] += rs[t + st]; rq[t] += rq[t + st]; }
    __syncthreads();
  }
  if (t == 0) {
    const float cnt = (float)B_ * 4096.f;
    float mean = rs[0] / cnt;
    float var  = rq[0] / cnt - mean * mean;
    float sc   = g_sim[ch] * rsqrtf(var + EPS_);
    ssc[ch] = sc;
    ssh[ch] = b_sim[ch] - mean * sc;
  }
}

// ---- kernel 8: attention core: sim -> softmax -> sv (WMMA) + sve ----------
__global__ void k_attn(const float* __restrict__ qkv,
                       const float* __restrict__ qsc, const float* __restrict__ qsh,
                       const float* __restrict__ emb,
                       const float* __restrict__ sig,
                       const float* __restrict__ ssc, const float* __restrict__ ssh,
                       float* __restrict__ outp) {
  int bg = blockIdx.x;
  int g = bg & 7, b = bg >> 3;
  int tid = threadIdx.x;          // 256 threads
  __shared__ float ql[8][64], kl[8][64], vl[16][64];
  __shared__ float sim[64][65];   // pad to spread LDS banks

  for (int idx = tid; idx < 2048; idx += 256) {
    int c = idx >> 6, hh = idx & 63;
    int o = g * 32 + c;
    float v = qsc[o] * qkv[((size_t)b * 256 + o) * 64 + hh] + qsh[o];
    if (c < 8)       ql[c][hh] = v;
    else if (c < 16) kl[c - 8][hh] = v;
    else             vl[c - 16][hh] = v;
  }
  __syncthreads();

  int wid = tid >> 5, lane = tid & 31;
  int row = lane & 15, kb = (lane >> 4) << 1;
  float s0 = ssc[g], t0 = ssh[g];
  for (int tt = 0; tt < 2; ++tt) {
    int tile = wid + tt * 8;
    int i0 = (tile >> 2) * 16, j0 = (tile & 3) * 16;
    v8f acc = {0.f, 0.f, 0.f, 0.f, 0.f, 0.f, 0.f, 0.f};
    for (int ks = 0; ks < 8; ks += 4) {
      v2f a, bb;
      a.x  = ql[ks + kb + 0][i0 + row];
      a.y  = ql[ks + kb + 1][i0 + row];
      bb.x = kl[ks + kb + 0][j0 + row];
      bb.y = kl[ks + kb + 1][j0 + row];
      acc = wmma4(a, bb, acc);
    }
    int col = j0 + (lane & 15);
    int mb  = (lane >> 4) * 8;
#pragma unroll
    for (int r = 0; r < 8; ++r) sim[i0 + mb + r][col] = s0 * acc[r] + t0;
  }
  __syncthreads();

  float sig1 = sig[b * 4 + 0], sig2 = sig[b * 4 + 1];
  float sig3 = sig[b * 4 + 2], sig4 = sig[b * 4 + 3];
  float s1 = ssc[8 + g],  t1 = ssh[8 + g];
  float s2 = ssc[16 + g], t2 = ssh[16 + g];
  for (int p = tid; p < 4096; p += 256) {
    int i = p >> 6, j = p & 63;
    float aq = 0.f, ak = 0.f;
#pragma unroll
    for (int c = 0; c < 8; ++c) {
      aq = fmaf(ql[c][i], emb[((size_t)(c * 64 + i)) * 64 + j], aq);
      ak = fmaf(kl[c][j], emb[((size_t)((8 + c) * 64 + j)) * 64 + i], ak);
    }
    sim[i][j] += (s1 * (sig1 * aq) + t1) + (s2 * (sig2 * ak) + t2);
  }
  __syncthreads();

  if (tid < 64) {                 // row-wise softmax over j
    int i = tid;
    float m = -1e30f;
    for (int j = 0; j < 64; ++j) m = fmaxf(m, sim[i][j]);
    float s = 0.f;
    for (int j = 0; j < 64; ++j) { float e = expf(sim[i][j] - m); sim[i][j] = e; s += e; }
    float inv = 1.f / s;
    for (int j = 0; j < 64; ++j) sim[i][j] *= inv;
  }
  __syncthreads();

  if (wid < 4) {
    // sv[c,i] = sig3 * sum_j v[c,j]*sim[i,j]  -> WMMA, M=c(16), N=i(16/tile), K=j(64)
    int i0 = wid * 16;
    v8f acc = {0.f, 0.f, 0.f, 0.f, 0.f, 0.f, 0.f, 0.f};
    for (int k0 = 0; k0 < 64; k0 += 4) {
      v2f a, bb;
      a.x  = vl[row][k0 + kb + 0];
      a.y  = vl[row][k0 + kb + 1];
      bb.x = sim[i0 + row][k0 + kb + 0];
      bb.y = sim[i0 + row][k0 + kb + 1];
      acc = wmma4(a, bb, acc);
    }
    int i  = i0 + (lane & 15);
    int mb = (lane >> 4) * 8;
#pragma unroll
    for (int r = 0; r < 8; ++r) {
      int c = mb + r;
      int o = 2 * (g * 16 + c);   // even channels = sv
      outp[((size_t)b * 256 + o) * 64 + i] = sig3 * acc[r];
    }
  } else {
    // sve[c,i] = sig4 * sum_j sim[i,j]*v_emb[c,i,j]
    int lt = tid - 128;           // 128 threads
    for (int p = lt; p < 1024; p += 128) {
      int c = p >> 6, i = p & 63;
      const float* ve = emb + ((size_t)((16 + c) * 64 + i)) * 64;
      float a = 0.f;
      for (int j = 0; j < 64; ++j) a = fmaf(sim[i][j], ve[j], a);
      int o = 2 * (g * 16 + c) + 1;   // odd channels = sve
      outp[((size_t)b * 256 + o) * 64 + i] = sig4 * a;
    }
  }
}

// ---- kernel 11: final BN + pair-sum + transpose to (N,C,H,W) --------------
__global__ void k_final(const float* __restrict__ outp,
                        const float* __restrict__ osc, const float* __restrict__ osh,
                        float* __restrict__ y) {
  int blk = blockIdx.x;           // n*128 + cf
  int n = blk >> 7, cf = blk & 127;
  int w = threadIdx.x;            // 64 threads
  int b = n * 64 + w;
  int o0 = 2 * cf;
  const float* p0 = outp + ((size_t)b * 256 + o0) * 64;
  const float* p1 = p0 + 64;
  float s0 = osc[o0],     h0 = osh[o0];
  float s1 = osc[o0 + 1], h1 = osh[o0 + 1];
  float* yp = y + (size_t)blk * 4096 + w;
  for (int h = 0; h < 64; ++h)
    yp[h * 64] = (s0 * p0[h] + h0) + (s1 * p1[h] + h1);
}

// ---------------------------------------------------------------------------
extern "C" void kernel_launch(void* const* d_in, const int* in_sizes, int n_in,
                              void* d_out, int out_size, void* d_ws, size_t ws_size,
                              hipStream_t stream) {
  (void)in_sizes; (void)n_in; (void)out_size; (void)ws_size;
  const float* x     = (const float*)d_in[0];
  const float* w_qkv = (const float*)d_in[1];
  const float* g_qkv = (const float*)d_in[2];
  const float* b_qkv = (const float*)d_in[3];
  const float* g_sim = (const float*)d_in[4];
  const float* b_sim = (const float*)d_in[5];
  const float* g_out = (const float*)d_in[6];
  const float* b_out = (const float*)d_in[7];
  const float* w1    = (const float*)d_in[8];
  const float* b1    = (const float*)d_in[9];
  const float* w2    = (const float*)d_in[10];
  const float* b2    = (const float*)d_in[11];
  const float* rel   = (const float*)d_in[12];
  float* y = (float*)d_out;

  // workspace carve-up (floats); total ~33.9M floats (~136 MB)
  float* ws   = (float*)d_ws;
  float* qkv  = ws;                       // B*256*64  = 16,777,216
  float* outp = qkv  + (size_t)B_ * 256 * 64;  // 16,777,216
  float* emb  = outp + (size_t)B_ * 256 * 64;  // 32*64*64 = 131,072
  float* xn   = emb  + 32 * 64 * 64;           // B*128    = 131,072
  float* sig  = xn   + (size_t)B_ * 128;       // B*4
  float* part = sig  + (size_t)B_ * 4;         // 8192*6
  float* qsc  = part + 8192 * 6;               // 256 each below
  float* qsh  = qsc + 256;
  float* osc  = qsh + 256;
  float* osh  = osc + 256;
  float* ssc  = osh + 256;                     // 24 each
  float* ssh  = ssc + 24;

  k_colmean  <<<N_ * C_, 64, 0, stream>>>(x, xn);
  k_gate     <<<B_, 128, 0, stream>>>(xn, w1, b1, w2, b2, sig);
  k_emb      <<<32 * 64, 64, 0, stream>>>(rel, emb);
  k_qkv_gemm <<<8192, 256, 0, stream>>>(x, w_qkv, qkv);
  k_chan_stats<<<256, 256, 0, stream>>>(qkv, g_qkv, b_qkv, qsc, qsh);
  k_sim_stats<<<B_ * 8, 256, 0, stream>>>(qkv, qsc, qsh, emb, sig, part);
  k_sim_final<<<24, 256, 0, stream>>>(part, g_sim, b_sim, ssc, ssh);
  k_attn     <<<B_ * 8, 256, 0, stream>>>(qkv, qsc, qsh, emb, sig, ssc, ssh, outp);
  k_chan_stats<<<256, 256, 0, stream>>>(outp, g_out, b_out, osc, osh);
  k_final    <<<N_ * C_, 64, 0, stream>>>(outp, osc, osh, y);
}